// GT_MLPNet_58600533787228
// MI455X (gfx1250) — compile-verified
//
#include <hip/hip_runtime.h>
#include <hip/hip_bf16.h>

#define K_DECAY 0.25f
#define SPAN 4           // INFLUENCE_SPAN
#define DFULL 256        // D (fixed by problem: W1 is 256x128)
#define ROWS 16          // output time-rows per block

typedef float v2f __attribute__((ext_vector_type(2)));
typedef float v8f __attribute__((ext_vector_type(8)));

__device__ __forceinline__ int dev_lower_bound(const int* __restrict__ a, int n, int key) {
    int lo = 0, hi = n;
    while (lo < hi) {
        int mid = (lo + hi) >> 1;
        if (a[mid] < key) lo = mid + 1; else hi = mid;
    }
    return lo;
}

// ---------------------------------------------------------------------------
// Kernel 1: time-decayed scatter of event embeddings, atomic-free in global.
// Blocks own disjoint 16-row windows of infl[T, 256]; sorted event_dates are
// binary-searched for the contributing event range. Because dates are sorted,
// buckets are monotone: accumulate each bucket's run in a register and flush
// to LDS (4x ds_add_f32) only on bucket change (~1 flush per ~122 events).
// ---------------------------------------------------------------------------
__global__ __launch_bounds__(256) void scatter_kernel(
    const int* __restrict__ h, const int* __restrict__ dates,
    const float* __restrict__ embed, float* __restrict__ infl,
    const int* __restrict__ p_start, const int* __restrict__ p_end,
    const int* __restrict__ p_tg, int n_ev, int T)
{
    __shared__ float acc[ROWS][DFULL];
    const int tid = threadIdx.x;       // == channel d (blockDim.x == DFULL)
    const int r0  = blockIdx.x * ROWS;

    #pragma unroll
    for (int i = 0; i < ROWS; ++i) acc[i][tid] = 0.0f;

    const int startb = p_start[0];
    const int endb   = p_end[0];
    const int tg     = p_tg[0];

    // buckets contributing to rows [r0, r0+ROWS): bk in [startb+r0-3, startb+r0+ROWS-1]
    long bkmin = (long)startb + r0 - (SPAN - 1);
    long bkmax = (long)startb + r0 + ROWS - 1;
    if (bkmin < 1) bkmin = 1;
    const int date_lo = (int)((bkmin - 1) * (long)tg + 1);
    const int date_hi = (int)(bkmax * (long)tg);   // inclusive
    const int e_lo = dev_lower_bound(dates, n_ev, date_lo);
    const int e_hi = dev_lower_bound(dates, n_ev, date_hi + 1);

    __syncthreads();

    const int d = tid;

    // run-length accumulation state (bucket index is uniform across the block)
    int   cur_base = -(1 << 28);       // sentinel: flush guards reject it
    float run      = 0.0f;

    auto flush = [&](int base, float r) {
        const int bk = base + startb;
        const bool valid = (bk + (SPAN - 1) >= startb) && (bk + (SPAN - 1) <= endb);
        if (!valid || r == 0.0f) return;
        #pragma unroll
        for (int dis = 0; dis < SPAN; ++dis) {
            const int row = base + dis;
            if (row >= r0 && row < r0 + ROWS && row >= 0 && row < T) {
                atomicAdd(&acc[row - r0][d], (1.0f - K_DECAY * (float)dis) * r); // ds_add_f32
            }
        }
    };

    for (int e = e_lo; e < e_hi; ++e) {
        const int dte  = dates[e];                 // uniform -> scalar load
        const int bk   = (dte - 1) / tg + 1;
        const int base = bk - startb;
        if (base != cur_base) {                    // uniform branch (s_cbranch)
            flush(cur_base, run);
            cur_base = base;
            run = 0.0f;
        }
        const int hv = h[e];                       // uniform -> scalar load
        if (e + 1 < e_hi) {
            // CDNA5 prefetch of the next event's embedding row (global_prefetch_b8)
            __builtin_prefetch(&embed[(size_t)h[e + 1] * DFULL + d], 0, 1);
        }
        run += embed[(size_t)hv * DFULL + d];      // coalesced 1KB row gather
    }
    flush(cur_base, run);

    __syncthreads();

    #pragma unroll
    for (int i = 0; i < ROWS; ++i) {
        const int r = r0 + i;
        if (r < T) infl[(size_t)r * DFULL + d] = acc[i][d];
    }
}

// ---------------------------------------------------------------------------
// Kernel 2: fused (x*W0 + infl*W1 + b) -> [16x256] @ W1[256x128] via
// v_wmma_f32_16x16x4_f32 -> relu -> @ W2[128x1] -> out[T,1].
// 8 waves/block, each wave owns one 16-wide N-chunk of the hidden layer.
// ---------------------------------------------------------------------------
__global__ __launch_bounds__(256) void mlp_kernel(
    const float* __restrict__ x, const float* __restrict__ infl,
    const float* __restrict__ Wxh, const float* __restrict__ bxh,
    const float* __restrict__ W1, const float* __restrict__ b1,
    const float* __restrict__ W2, const float* __restrict__ b2,
    float* __restrict__ out, int L, int T, int Dh)
{
    __shared__ float fin[ROWS][DFULL];   // fused "final" tile, 16 KB
    __shared__ float osum[ROWS];

    const int tid = threadIdx.x;
    const int r0  = blockIdx.x * ROWS;

    if (tid < ROWS) osum[tid] = 0.0f;

    const float w0 = Wxh[0], w1 = Wxh[1], bx = bxh[0];

    // Stage final tile: thread -> (row = tid/16, 16 contiguous channels)
    {
        const int row = tid >> 4;
        const int c0  = (tid & 15) << 4;
        const int rg  = r0 + row;
        if (rg < T) {
            const float* ip = &infl[(size_t)rg * DFULL + c0];
            if (rg < L) {
                const float* xp = &x[(size_t)rg * DFULL + c0];
                #pragma unroll
                for (int c = 0; c < 16; ++c)
                    fin[row][c0 + c] = xp[c] * w0 + ip[c] * w1 + bx;
            } else {
                #pragma unroll
                for (int c = 0; c < 16; ++c)
                    fin[row][c0 + c] = ip[c];
            }
        } else {
            #pragma unroll
            for (int c = 0; c < 16; ++c)
                fin[row][c0 + c] = 0.0f;
        }
    }
    __syncthreads();

    const int wave = tid >> 5;          // 0..7 -> N-chunk
    const int lane = tid & 31;
    const int nloc = lane & 15;         // column within 16-wide chunk / row of A
    const int hi   = lane >> 4;         // selects K pair (0,1) vs (2,3)
    const int n    = (wave << 4) + nloc;

    v8f c = {0.f, 0.f, 0.f, 0.f, 0.f, 0.f, 0.f, 0.f};

    #pragma unroll 8
    for (int k = 0; k < DFULL; k += 4) {
        const int kk = k + hi * 2;
        v2f a, b;
        // A 16x4 f32: lanes 0-15 hold rows m=0..15 with K0,K1; lanes 16-31 K2,K3
        a.x = fin[nloc][kk];
        a.y = fin[nloc][kk + 1];
        // B 4x16 f32: lane holds column n, K rows (hi*2, hi*2+1)
        b.x = W1[(size_t)kk * Dh + n];
        b.y = W1[(size_t)(kk + 1) * Dh + n];
        c = __builtin_amdgcn_wmma_f32_16x16x4_f32(
                /*neg_a=*/false, a, /*neg_b=*/false, b,
                /*c_mod=*/(short)0, c, /*reuse_a=*/false, /*reuse_b=*/false);
    }

    // Readout: hid = relu(c + b1[n]);  out_m += hid * W2[n]
    const float bias = b1[n];
    const float w2v  = W2[n];
    #pragma unroll
    for (int j = 0; j < 8; ++j) {
        const int m = j + 8 * hi;       // C/D layout: lane<16 -> m=j, lane>=16 -> m=j+8
        float hval = c[j] + bias;
        hval = hval > 0.0f ? hval : 0.0f;
        atomicAdd(&osum[m], hval * w2v);   // ds_add_f32 reduction across 16 lanes x 8 waves
    }
    __syncthreads();

    if (tid < ROWS) {
        const int r = r0 + tid;
        if (r < T) out[r] = osum[tid] + b2[0];
    }
}

extern "C" void kernel_launch(void* const* d_in, const int* in_sizes, int n_in,
                              void* d_out, int out_size, void* d_ws, size_t ws_size,
                              hipStream_t stream) {
    const int*   h     = (const int*)d_in[0];
    const float* x     = (const float*)d_in[1];
    const int*   dates = (const int*)d_in[2];
    const float* embed = (const float*)d_in[3];
    const float* Wxh   = (const float*)d_in[4];
    const float* bxh   = (const float*)d_in[5];
    const float* W1    = (const float*)d_in[6];
    const float* b1    = (const float*)d_in[7];
    const float* W2    = (const float*)d_in[8];
    const float* b2    = (const float*)d_in[9];
    const int* p_start = (const int*)d_in[10];
    const int* p_end   = (const int*)d_in[11];
    const int* p_tg    = (const int*)d_in[12];

    const int n_ev = in_sizes[0];
    const int Dh   = in_sizes[7];        // 128
    const int D    = 2 * Dh;             // 256 (== DFULL)
    const int L    = in_sizes[1] / D;    // 2048
    const int T    = out_size;           // 4096
    float* infl    = (float*)d_ws;       // [T, D] f32 scratch (4 MB)

    const int nblocks = (T + ROWS - 1) / ROWS;
    scatter_kernel<<<nblocks, DFULL, 0, stream>>>(h, dates, embed, infl,
                                                  p_start, p_end, p_tg, n_ev, T);
    mlp_kernel<<<nblocks, DFULL, 0, stream>>>(x, infl, Wxh, bxh,
                                              W1, b1, W2, b2,
                                              (float*)d_out, L, T, Dh);
}